// LocalConnectivity_58179626992077
// MI455X (gfx1250) — compile-verified
//
#include <hip/hip_runtime.h>

// ---------------------------------------------------------------------------
// Diamond stencil (Manhattan radius 5, weights e^{-d}) on a 64x1024x1024 torus.
//
// Factorization: t_r[i,j] = sum_{|dy|<=r} e^{-|dy|} f[i,j+dy]  (nested)
//   out[i,j] = (t5 - f)[i,j] + sum_{a=1..5} e^{-a} (t_{5-a}[i-a,j] + t_{5-a}[i+a,j])
// Vertical combine via a rolling 11-register accumulator (scatter per input row).
//
// CDNA5 specifics:
//  - double-buffered GLOBAL_LOAD_ASYNC_TO_LDS_B32 staging (ASYNCcnt), partial
//    s_wait_asynccnt keeps the prefetched chunk in flight across the barrier;
//    every wave issues exactly 48 async ops per chunk (uniform EXEC) so the
//    partial-wait bookkeeping is exact
//  - packed fp32 (V_PK_FMA_F32 / V_PK_ADD_F32): each thread owns two adjacent
//    columns as a float2; LDS reads become b64
//  - non-temporal output stores (output is never re-read; keep L2 for halos)
// No WMMA by design: fp32 WMMA is 16x16x4 only; an 11-tap band matmul wastes
// >61% of K slots.
// ---------------------------------------------------------------------------

#define HDIM   1024
#define WDIM   1024
#define NTHR   128          // 4 waves
#define TW     256          // output tile width  (2 columns per thread)
#define TR     256          // output tile height
#define HALO   5
#define SW     272          // staged width: 256 + 2*5 = 266, padded to 272
#define CH     16           // staged rows per chunk
#define NASYNC (3 * CH)                          // 48 async b32 issues / wave / chunk
#define NCHUNKS ((TR + 2 * HALO + CH - 1) / CH)  // 17 (last chunk over-stages; harmless)

typedef float v2f __attribute__((ext_vector_type(2)));
typedef __attribute__((address_space(1))) int as1_int;   // global-AS int*
typedef __attribute__((address_space(3))) int as3_int;   // LDS-AS int*

__device__ __forceinline__ void async_copy_f32(const float* __restrict__ gsrc,
                                               float* lds_dst) {
#if defined(__gfx1250__) && __has_builtin(__builtin_amdgcn_global_load_async_to_lds_b32)
  __builtin_amdgcn_global_load_async_to_lds_b32(
      (as1_int*)(unsigned long long)(const void*)gsrc,
      (as3_int*)(unsigned)(unsigned long long)(const void*)lds_dst,
      0, 0);
#else
  asm volatile("global_load_async_to_lds_b32 %0, %1, off"
               :
               : "v"((unsigned)(unsigned long long)(const void*)lds_dst),
                 "v"((unsigned long long)(const void*)gsrc)
               : "memory");
#endif
}

__device__ __forceinline__ void wait_async_keep_next() {
#if defined(__gfx1250__) && __has_builtin(__builtin_amdgcn_s_wait_asynccnt)
  __builtin_amdgcn_s_wait_asynccnt(NASYNC);   // leave the prefetched chunk in flight
#else
  asm volatile("s_wait_asynccnt %0" :: "n"(NASYNC) : "memory");
#endif
}

__device__ __forceinline__ void wait_async_all() {
#if defined(__gfx1250__) && __has_builtin(__builtin_amdgcn_s_wait_asynccnt)
  __builtin_amdgcn_s_wait_asynccnt(0);
#else
  asm volatile("s_wait_asynccnt 0" ::: "memory");
#endif
}

__device__ __forceinline__ v2f pk_fma(float w, v2f x, v2f c) {
  v2f wv; wv.x = w; wv.y = w;
  return __builtin_elementwise_fma(wv, x, c);   // -> v_pk_fma_f32
}

__global__ __launch_bounds__(NTHR, 4)
void LocalConnectivity_diamond_kernel(const float* __restrict__ src,
                                      float* __restrict__ dst) {
  __shared__ float sbuf[2][CH * SW];   // 2 * 16 * 272 * 4 B = 34 KiB of 320 KiB

  const int tid = threadIdx.x;
  const int c0  = blockIdx.x * TW;
  const int r0  = blockIdx.y * TR;
  const unsigned planeBase = (unsigned)blockIdx.z * (HDIM * WDIM);

  // Loop-invariant wrapped column offsets for this thread's 3 staging slots.
  const int gcol0 = (c0 - HALO + tid) & (WDIM - 1);
  const int gcol1 = (c0 - HALO + tid + NTHR) & (WDIM - 1);
  const int tail  = 2 * NTHR + (tid & 15);              // 256..271 (8x redundant)
  const int gcol2 = (c0 - HALO + tail) & (WDIM - 1);

  // Stage chunk `ck` (staged-local rows ck*CH .. ck*CH+15) into sbuf[ck&1].
  // Scalar row base (SALU) + per-lane column offset: GVS-form async copies.
  auto stage = [&](int ck) {
    const int rowBase = ck * CH;
    float* buf = sbuf[ck & 1];
#pragma unroll
    for (int rr = 0; rr < CH; ++rr) {
      const int grow = (r0 - HALO + rowBase + rr) & (HDIM - 1);
      const float* srow = src + planeBase + (unsigned)grow * WDIM;
      float* drow = buf + rr * SW;
      async_copy_f32(srow + gcol0, drow + tid);
      async_copy_f32(srow + gcol1, drow + tid + NTHR);
      async_copy_f32(srow + gcol2, drow + tail);
    }
  };

  stage(0);
  stage(1);

  // exp(-d) ring weights
  constexpr float W1 = 0.36787944117144233f;
  constexpr float W2 = 0.13533528323661270f;
  constexpr float W3 = 0.049787068367863944f;
  constexpr float W4 = 0.018315638888734179f;
  constexpr float W5 = 0.0067379469990854670f;

  v2f acc[11];
#pragma unroll
  for (int j = 0; j < 11; ++j) { acc[j].x = 0.f; acc[j].y = 0.f; }

  // this thread's output column pair: c0 + 2*tid, c0 + 2*tid + 1
  const unsigned outColBase =
      planeBase + (unsigned)r0 * WDIM + (unsigned)(c0 + 2 * tid);

  for (int ck = 0; ck < NCHUNKS; ++ck) {
    if (ck + 1 < NCHUNKS) wait_async_keep_next();   // chunk ck done; ck+1 in flight
    else                  wait_async_all();
    __syncthreads();

    const float* buf = sbuf[ck & 1];
#pragma unroll
    for (int r = 0; r < CH; ++r) {
      const int lr = ck * CH + r;                        // staged-local row, 0..271
      // staged cols for this pair: sc = 2*tid .. 2*tid+11, 8B-aligned pairs
      const v2f* row2 = (const v2f*)(buf + r * SW) + tid;
      const v2f q0 = row2[0];
      const v2f q1 = row2[1];
      const v2f q2 = row2[2];
      const v2f q3 = row2[3];
      const v2f q4 = row2[4];
      const v2f q5 = row2[5];

      // shifted pairs s_m = (f[2t+m], f[2t+m+1]); odd m via component swizzle
      v2f F;  F.x  = q2.y; F.y  = q3.x;    // s5: center taps for (colA, colB)
      v2f s3; s3.x = q1.y; s3.y = q2.x;
      v2f s7; s7.x = q3.y; s7.y = q4.x;
      v2f s1; s1.x = q0.y; s1.y = q1.x;
      v2f s9; s9.x = q4.y; s9.y = q5.x;

      const v2f p1 = q2 + q3;   // s4 + s6
      const v2f p2 = s3 + s7;
      const v2f p3 = q1 + q4;   // s2 + s8
      const v2f p4 = s1 + s9;
      const v2f p5 = q0 + q5;   // s0 + s10

      const v2f t1 = pk_fma(W1, p1, F);
      const v2f t2 = pk_fma(W2, p2, t1);
      const v2f t3 = pk_fma(W3, p3, t2);
      const v2f t4 = pk_fma(W4, p4, t3);
      const v2f t5 = pk_fma(W5, p5, t4);

      // scatter this input row into the 11 in-flight output rows
      acc[5] += (t5 - F);                                 // dx = 0 (center excluded)
      acc[4]  = pk_fma(W1, t4, acc[4]);  acc[6]  = pk_fma(W1, t4, acc[6]);   // |dx|=1
      acc[3]  = pk_fma(W2, t3, acc[3]);  acc[7]  = pk_fma(W2, t3, acc[7]);   // |dx|=2
      acc[2]  = pk_fma(W3, t2, acc[2]);  acc[8]  = pk_fma(W3, t2, acc[8]);   // |dx|=3
      acc[1]  = pk_fma(W4, t1, acc[1]);  acc[9]  = pk_fma(W4, t1, acc[9]);   // |dx|=4
      acc[0]  = pk_fma(W5, F,  acc[0]);  acc[10] = pk_fma(W5, F,  acc[10]);  // |dx|=5

      // acc[0] corresponds to output row (lr - 10); complete after this row.
      if (lr >= 2 * HALO && lr <= TR + 2 * HALO - 1) {
        __builtin_nontemporal_store(
            acc[0], (v2f*)(dst + outColBase + (unsigned)(lr - 2 * HALO) * WDIM));
      }

      // rotate the accumulator ring (renamed away by the unroller)
#pragma unroll
      for (int j = 0; j < 10; ++j) acc[j] = acc[j + 1];
      acc[10].x = 0.f; acc[10].y = 0.f;
    }

    __syncthreads();                     // everyone done reading sbuf[ck&1]
    if (ck + 2 < NCHUNKS) stage(ck + 2); // refill it for chunk ck+2
  }
}

extern "C" void kernel_launch(void* const* d_in, const int* in_sizes, int n_in,
                              void* d_out, int out_size, void* d_ws, size_t ws_size,
                              hipStream_t stream) {
  const float* src = (const float*)d_in[0];
  float* dst = (float*)d_out;
  const int B = in_sizes[0] / (HDIM * WDIM);   // 64
  dim3 grid(WDIM / TW, HDIM / TR, B);
  dim3 block(NTHR);
  hipLaunchKernelGGL(LocalConnectivity_diamond_kernel, grid, block, 0, stream,
                     src, dst);
}